// BilinearInterpolate_77386720740115
// MI455X (gfx1250) — compile-verified
//
#include <hip/hip_runtime.h>
#include <hip/hip_bf16.h>
#include <stdint.h>

// Bilinear grid-sample, MI455X (gfx1250).
// Im: [B=16, C=32, H=256, W=256] f32, G: [B, 2, OH=256, OW=256] f32 pixel coords,
// out: [B, C, OH, OW] f32.
//
// Memory-bound: HBM floor ~277 MB (~12us @ 23.3 TB/s). Im (134 MB) fits in the
// 192 MB L2 -> gathers are L2 hits after first touch. Output stored with
// non-temporal hint so writes don't evict the image. Coordinates are streamed
// through LDS with gfx1250 async-to-LDS loads (ASYNCcnt double-buffer pipeline).

#define BLK    256
#define NTILES 4
#define HW     65536      // 256*256
#define NCH    32

__device__ __forceinline__ void async_copy_b32(uint32_t lds_byte_off, const float* gaddr) {
    // GLOBAL_LOAD_ASYNC_TO_LDS_B32: each lane copies 4 bytes from the global address in vaddr to the LDS address in vdst; tracked by ASYNCcnt.
    asm volatile("global_load_async_to_lds_b32 %0, %1, off"
                 :: "v"(lds_byte_off), "v"((unsigned long long)(uintptr_t)gaddr)
                 : "memory");
}

__global__ __launch_bounds__(BLK) void BilinearInterpolate_77386720740115_kernel(
    const float* __restrict__ Im, const float* __restrict__ G, float* __restrict__ out)
{
    __shared__ float scoord[2][2][BLK];   // [buffer][gx/gy][lane]

    const int tid           = threadIdx.x;
    const int blocksPerImg  = HW / (BLK * NTILES);          // 64
    const int b             = blockIdx.x / blocksPerImg;    // batch
    const int blkInImg      = blockIdx.x % blocksPerImg;
    const int pixBase       = blkInImg * (BLK * NTILES);

    const float* __restrict__ gxBase = G + (size_t)b * 2 * HW + pixBase;
    const float* __restrict__ gyBase = gxBase + HW;
    const float* __restrict__ ImBase = Im + (size_t)b * NCH * HW;
    float*       __restrict__ outBase = out + (size_t)b * NCH * HW + pixBase;

    // ---- prologue: async-stage tile 0 coords into LDS buffer 0 ----
    async_copy_b32((uint32_t)(uintptr_t)&scoord[0][0][tid], gxBase + tid);
    async_copy_b32((uint32_t)(uintptr_t)&scoord[0][1][tid], gyBase + tid);

    for (int t = 0; t < NTILES; ++t) {
        const int cur = t & 1;
        if (t + 1 < NTILES) {
            const int nxt = cur ^ 1;
            async_copy_b32((uint32_t)(uintptr_t)&scoord[nxt][0][tid],
                           gxBase + (t + 1) * BLK + tid);
            async_copy_b32((uint32_t)(uintptr_t)&scoord[nxt][1][tid],
                           gyBase + (t + 1) * BLK + tid);
            // 2 newly issued ops may be pending; <=2 guarantees tile t's pair landed
            asm volatile("s_wait_asynccnt 2" ::: "memory");
        } else {
            asm volatile("s_wait_asynccnt 0" ::: "memory");
        }
        __syncthreads();   // all waves' async writes for tile t visible in LDS

        const float gx = scoord[cur][0][tid];
        const float gy = scoord[cur][1][tid];

        // Replicate-pad bilinear: xi0=floor(gx) clamped to [0,254]; edge case
        // folds into fx (fx -> 1 selects column xi0+1 == 255), matching reference.
        int xi = (int)floorf(gx);
        int yi = (int)floorf(gy);
        xi = xi < 0 ? 0 : (xi > 254 ? 254 : xi);
        yi = yi < 0 ? 0 : (yi > 254 ? 254 : yi);
        const float fx = gx - (float)xi;
        const float fy = gy - (float)yi;

        const int r0 = yi * 256 + xi;     // top-left tap within one channel plane
        const int r1 = r0 + 256;          // bottom-left tap
        const int q  = t * BLK + tid;     // pixel offset within this image plane

        #pragma unroll 8
        for (int c = 0; c < NCH; ++c) {
            const float* __restrict__ pc = ImBase + ((size_t)c << 16);
            const float v00 = pc[r0];
            const float v01 = pc[r0 + 1];
            const float v10 = pc[r1];
            const float v11 = pc[r1 + 1];
            const float top = fmaf(fx, v01 - v00, v00);
            const float bot = fmaf(fx, v11 - v10, v10);
            const float res = fmaf(fy, bot - top, top);
            // NT store: don't let 134 MB of output evict the L2-resident image
            __builtin_nontemporal_store(res, outBase + q + ((size_t)c << 16));
        }
        __syncthreads();   // tile t fully consumed before buffer reuse at t+2
    }
}

extern "C" void kernel_launch(void* const* d_in, const int* in_sizes, int n_in,
                              void* d_out, int out_size, void* d_ws, size_t ws_size,
                              hipStream_t stream) {
    const float* Im = (const float*)d_in[0];   // [16,32,256,256]
    const float* G  = (const float*)d_in[1];   // [16,2,256,256]
    float* out      = (float*)d_out;           // [16,32,256,256]

    const int B = 16;
    const int blocksPerImg = HW / (BLK * NTILES);   // 64
    dim3 grid(B * blocksPerImg);                    // 1024 blocks
    dim3 block(BLK);                                // 256 threads = 8 wave32s
    BilinearInterpolate_77386720740115_kernel<<<grid, block, 0, stream>>>(Im, G, out);
}